// DeepRecursiveNLM_40432822124589
// MI455X (gfx1250) — compile-verified
//
#include <hip/hip_runtime.h>

typedef float v2f __attribute__((ext_vector_type(2)));
typedef float v4f __attribute__((ext_vector_type(4)));
typedef float v8f __attribute__((ext_vector_type(8)));

#define B_SZ    32768
#define D_SZ    128
#define HIST_SZ 30
#define M_SZ    15
#define H_SZ    4
#define W1_N    (M_SZ*H_SZ*D_SZ)   // 7680

// ---------------------------------------------------------------------------
// Prep: W2 = T1 @ W1c, W3 = T2 @ W2 where T = alpha*U*diag(f(S)/S)*U^T + r*I,
// obtained from a double-precision Jacobi eigensolve of the 4x4 Gram matrix.
// ---------------------------------------------------------------------------
__device__ static void jacobi_spectral_T(const float W[H_SZ][D_SZ],
                                         const float* sscale, const float* sshift,
                                         double alpha, double rres,
                                         double T[H_SZ][H_SZ])
{
    double G[4][4];
    for (int i = 0; i < 4; ++i)
        for (int j = 0; j < 4; ++j) {
            double s = 0.0;
            for (int d = 0; d < D_SZ; ++d) s += (double)W[i][d] * (double)W[j][d];
            G[i][j] = s;
        }
    double V[4][4] = {{1,0,0,0},{0,1,0,0},{0,0,1,0},{0,0,0,1}};
    for (int sweep = 0; sweep < 30; ++sweep) {
        for (int p = 0; p < 3; ++p) for (int q = p + 1; q < 4; ++q) {
            double apq = G[p][q];
            if (fabs(apq) < 1e-300) continue;
            double theta = (G[q][q] - G[p][p]) / (2.0 * apq);
            double t = ((theta >= 0.0) ? 1.0 : -1.0) / (fabs(theta) + sqrt(theta*theta + 1.0));
            double c = 1.0 / sqrt(t*t + 1.0), s = t * c;
            for (int k = 0; k < 4; ++k) {
                double gkp = G[k][p], gkq = G[k][q];
                G[k][p] = c*gkp - s*gkq; G[k][q] = s*gkp + c*gkq;
            }
            for (int k = 0; k < 4; ++k) {
                double gpk = G[p][k], gqk = G[q][k];
                G[p][k] = c*gpk - s*gqk; G[q][k] = s*gpk + c*gqk;
            }
            for (int k = 0; k < 4; ++k) {
                double vkp = V[k][p], vkq = V[k][q];
                V[k][p] = c*vkp - s*vkq; V[k][q] = s*vkp + c*vkq;
            }
        }
    }
    double lam[4]; int idx[4];
    for (int i = 0; i < 4; ++i) { lam[i] = G[i][i]; idx[i] = i; }
    for (int i = 0; i < 3; ++i)                      // sort descending (SVD order)
        for (int j = i + 1; j < 4; ++j)
            if (lam[idx[j]] > lam[idx[i]]) { int t2 = idx[i]; idx[i] = idx[j]; idx[j] = t2; }
    double ratio[4];
    for (int i = 0; i < 4; ++i) {
        double lv = lam[idx[i]]; if (lv < 0.0) lv = 0.0;
        double S = sqrt(lv);
        double x = (double)sscale[i];
        double sp = (x > 30.0) ? x : log1p(exp(x));  // softplus
        double Snew = sp * S + (double)sshift[i];
        ratio[i] = Snew / ((S > 1e-30) ? S : 1e-30);
    }
    for (int p = 0; p < 4; ++p)
        for (int q = 0; q < 4; ++q) {
            double s = 0.0;
            for (int i = 0; i < 4; ++i) s += V[p][idx[i]] * ratio[i] * V[q][idx[i]];
            T[p][q] = alpha * s + ((p == q) ? rres : 0.0);
        }
}

__global__ void __launch_bounds__(128)
prep_kernel(const float* __restrict__ w1, const float* __restrict__ cw,
            const float* __restrict__ sscale, const float* __restrict__ sshift,
            const float* __restrict__ alpha_p, const float* __restrict__ rres_p,
            float* __restrict__ W2W3)
{
    __shared__ float  sWa[H_SZ][D_SZ];
    __shared__ float  sWb[H_SZ][D_SZ];
    __shared__ double sT[H_SZ][H_SZ];
    const int t = threadIdx.x;

    double wm[M_SZ];
    {
        double mx = -1e300;
        for (int m = 0; m < M_SZ; ++m) { double v = (double)cw[m]; if (v > mx) mx = v; }
        double s = 0.0;
        for (int m = 0; m < M_SZ; ++m) { wm[m] = exp((double)cw[m] - mx); s += wm[m]; }
        for (int m = 0; m < M_SZ; ++m) wm[m] /= s;
    }
    const double a = (double)alpha_p[0], r = (double)rres_p[0];

    if (t < D_SZ)
        for (int h = 0; h < H_SZ; ++h) {
            double s = 0.0;
            for (int m = 0; m < M_SZ; ++m) s += wm[m] * (double)w1[(m*H_SZ + h)*D_SZ + t];
            sWa[h][t] = (float)s;
        }
    __syncthreads();
    if (t == 0) jacobi_spectral_T(sWa, sscale, sshift, a, r, sT);
    __syncthreads();
    if (t < D_SZ)
        for (int h = 0; h < H_SZ; ++h) {
            double s = 0.0;
            for (int j = 0; j < H_SZ; ++j) s += sT[h][j] * (double)sWa[j][t];
            sWb[h][t] = (float)s;
            W2W3[h*D_SZ + t] = (float)s;                     // W2
        }
    __syncthreads();
    if (t == 0) jacobi_spectral_T(sWb, sscale, sshift, a, r, sT);
    __syncthreads();
    if (t < D_SZ)
        for (int h = 0; h < H_SZ; ++h) {
            double s = 0.0;
            for (int j = 0; j < H_SZ; ++j) s += sT[h][j] * (double)sWb[j][t];
            W2W3[(H_SZ + h)*D_SZ + t] = (float)s;            // W3
        }
}

// ---------------------------------------------------------------------------
// Branch-free group-of-4 lane sum via DPP quad_perm (pure VALU, no LDS).
// ---------------------------------------------------------------------------
__device__ static inline float dpp_sum4(float x)
{
    int t1 = __builtin_amdgcn_update_dpp(0, __float_as_int(x),
                                         0xB1 /*quad_perm(1,0,3,2)*/, 0xF, 0xF, true);
    float s = x + __int_as_float(t1);
    int t2 = __builtin_amdgcn_update_dpp(0, __float_as_int(s),
                                         0x4E /*quad_perm(2,3,0,1)*/, 0xF, 0xF, true);
    return s + __int_as_float(t2);
}

// ---------------------------------------------------------------------------
// Main: HBM-bound stream. Per wave: 16 b-rows x one d; h1 via 4 chained
// V_WMMA_F32_16X16X4_F32 (K=15 zero-padded to 16). All LDS accesses are
// unconditional with cndmask-selected addresses into zero-pad regions, so the
// hot loop has no exec-mask branching; h-reduction is DPP-only.
// ---------------------------------------------------------------------------
__global__ void __launch_bounds__(256, 2)
main_kernel(const float* __restrict__ pre,
            const float* __restrict__ w1,
            const float* __restrict__ b1g,
            const float* __restrict__ b2g,
            const float* __restrict__ b3g,
            const float* __restrict__ W2W3,
            float* __restrict__ out)
{
    __shared__ float lds_w1z[W1_N + 16];     // weights_1 + 16-float zero pad
    __shared__ v4f   lds_p[H_SZ*D_SZ + 1];   // packed {b1,b2,W2,W3} + zero pad
    __shared__ float lds_b3[D_SZ];
    __shared__ float lds_outT[8][16][16];    // [wave][d-col][b-row]
    __shared__ float lds_dummy[32 * 8];      // junk sink for masked stores

    const int tid = threadIdx.x;
    for (int i = tid; i < W1_N; i += 256) lds_w1z[i] = w1[i];
    if (tid < 16) lds_w1z[W1_N + tid] = 0.0f;
    for (int i = tid; i < H_SZ*D_SZ; i += 256)
        lds_p[i] = (v4f){ b1g[i], b2g[i], W2W3[i], W2W3[H_SZ*D_SZ + i] };
    if (tid == 0) lds_p[H_SZ*D_SZ] = (v4f){0.f, 0.f, 0.f, 0.f};
    if (tid < D_SZ) lds_b3[tid] = b3g[tid];
    __syncthreads();

    const int  wave  = tid >> 5;
    const int  lane  = tid & 31;
    const int  n16   = lane & 15;            // A: row M; B/D: column N
    const int  hi    = lane >> 4;            // K half select
    const bool hval  = (n16 < H_SZ);
    const int  hoff  = n16 * D_SZ;           // valid only when hval
    const int  b0    = blockIdx.x * 16;
    const int  row   = b0 + n16;
    const int  dbase = wave * 16;

    int base = (row * D_SZ + dbase) * HIST_SZ + (HIST_SZ - M_SZ);

    for (int dd = 0; dd < 16; ++dd, base += HIST_SZ) {
        const int d = dbase + dd;
        if (dd < 15) __builtin_prefetch(&pre[base + HIST_SZ], 0, 0);
        const int wbase = hoff + d;          // lds_w1z column base for this lane

        v2f afr[4], bfr[4];
        #pragma unroll
        for (int c = 0; c < 4; ++c) {
            const int m0  = c*4 + hi*2;      // this lane-half owns K = m0, m0+1
            const int m1  = m0 + 1;
            const int m1c = (m1 > M_SZ - 1) ? (M_SZ - 1) : m1;   // clamp addr
            float x0  = __builtin_nontemporal_load(&pre[base + m0]);
            float x1l = __builtin_nontemporal_load(&pre[base + m1c]);
            float x1  = (m1 < M_SZ) ? x1l : 0.0f;                 // VALU select
            afr[c] = (v2f){x0, x1};
            // unconditional LDS loads; invalid lanes/taps read the zero pad
            const int i0 = hval              ? (m0 * H_SZ * D_SZ + wbase) : W1_N;
            const int i1 = (hval && m1 < M_SZ) ? (m1 * H_SZ * D_SZ + wbase) : W1_N;
            bfr[c] = (v2f){ lds_w1z[i0], lds_w1z[i1] };
        }

        v8f acc = {};
        #pragma unroll
        for (int c = 0; c < 4; ++c)
            acc = __builtin_amdgcn_wmma_f32_16x16x4_f32(
                      false, afr[c], false, bfr[c], (short)0, acc, false, false);

        // packed params; zero pad for lanes n16>=4 -> whole chain yields z=0
        const v4f P = lds_p[hval ? (hoff + d) : (H_SZ*D_SZ)];
        const float b1s = P.x, b2s = P.y, w2s = P.z, w3s = P.w;

        float zout[8];
        #pragma unroll
        for (int i = 0; i < 8; ++i) {        // D VGPR i: row = b0 + i + 8*hi
            float h1v = fmaxf(acc[i] + b1s, 0.0f);
            float h2v = fmaxf(fmaf(h1v, w2s, b2s), 0.0f);
            zout[i] = dpp_sum4(h2v * w3s);   // lanes 0..3 / 16..19 hold row sum
        }
        // two packed 128-bit stores; non-writer lanes steered to dummy slots
        float* wp = (n16 == 0) ? &lds_outT[wave][dd][hi * 8]
                               : &lds_dummy[lane * 8];
        *(v4f*)(wp + 0) = (v4f){zout[0], zout[1], zout[2], zout[3]};
        *(v4f*)(wp + 4) = (v4f){zout[4], zout[5], zout[6], zout[7]};
    }
    __syncthreads();

    #pragma unroll
    for (int r2 = 0; r2 < 8; ++r2) {
        const int rr = r2*2 + hi;
        const float v = lds_outT[wave][n16][rr] + lds_b3[dbase + n16];
        __builtin_nontemporal_store(v, &out[(b0 + rr)*D_SZ + dbase + n16]);
    }
}

// ---------------------------------------------------------------------------
extern "C" void kernel_launch(void* const* d_in, const int* in_sizes, int n_in,
                              void* d_out, int out_size, void* d_ws, size_t ws_size,
                              hipStream_t stream)
{
    (void)in_sizes; (void)n_in; (void)out_size; (void)ws_size;
    const float* pre = (const float*)d_in[0];
    const float* w1  = (const float*)d_in[1];
    const float* b1  = (const float*)d_in[2];
    const float* b2  = (const float*)d_in[3];
    const float* b3  = (const float*)d_in[4];
    const float* cw  = (const float*)d_in[5];
    const float* ss  = (const float*)d_in[6];
    const float* sh  = (const float*)d_in[7];
    const float* al  = (const float*)d_in[8];
    const float* rs  = (const float*)d_in[9];
    float* out  = (float*)d_out;
    float* W2W3 = (float*)d_ws;   // [0,512): W2 (h*128+d), [512,1024): W3

    prep_kernel<<<1, 128, 0, stream>>>(w1, cw, ss, sh, al, rs, W2W3);
    main_kernel<<<B_SZ / 16, 256, 0, stream>>>(pre, w1, b1, b2, b3, W2W3, out);
}